// PodActor_46248207843537
// MI455X (gfx1250) — compile-verified
//
#include <hip/hip_runtime.h>
#include <hip/hip_bf16.h>
#include <math.h>

// ---------------------------------------------------------------------------
// PodActor forward for MI455X (gfx1250, wave32), fp32 WMMA 16x16x4.
// Weights are pre-packed (per call) into WMMA B-fragment order so the hot
// loops are pure {ds_load_b64 A, global_load_b64 B, v_wmma} with no guards.
// ---------------------------------------------------------------------------

typedef float v2f __attribute__((ext_vector_type(2)));
typedef float v8f __attribute__((ext_vector_type(8)));

#define NB  1024
#define NS  256
#define NBS (NB * NS)

#define PX 100   // LDS pitches (floats): even (b64-aligned), not mult. of 64
#define PY 100
#define PC 100

__device__ inline v8f wmma4(v2f a, v2f b, v8f c) {
  return __builtin_amdgcn_wmma_f32_16x16x4_f32(false, a, false, b, (short)0, c,
                                               false, false);
}
__device__ inline float sigm(float x) { return 1.f / (1.f + expf(-x)); }
__device__ inline v2f ld2(const float* p) { return *(const v2f*)p; }

// ---- packed-weight layout offsets (floats): [bias O][frags NT*KT*64] -------
constexpr int lsz(int I, int O) { return O + (O / 16) * ((I + 3) / 4) * 64; }
constexpr int OF_PE_RUN1 = 0;
constexpr int OF_PE_RUN2 = OF_PE_RUN1 + lsz(25, 64);
constexpr int OF_PE_BLK1 = OF_PE_RUN2 + lsz(64, 32);
constexpr int OF_PE_BLK2 = OF_PE_BLK1 + lsz(25, 64);
constexpr int OF_QKV     = OF_PE_BLK2 + lsz(64, 32);
constexpr int OF_AOUT    = OF_QKV + lsz(32, 96);
constexpr int OF_FF1     = OF_AOUT + lsz(32, 32);
constexpr int OF_FF2     = OF_FF1 + lsz(32, 64);
constexpr int OF_TM1     = OF_FF2 + lsz(64, 32);
constexpr int OF_TM2     = OF_TM1 + lsz(14, 32);
constexpr int OF_ENR1    = OF_TM2 + lsz(32, 16);
constexpr int OF_ENR2    = OF_ENR1 + lsz(14, 32);
constexpr int OF_ENB1    = OF_ENR2 + lsz(32, 16);
constexpr int OF_ENB2    = OF_ENB1 + lsz(14, 32);
constexpr int OF_BBR1    = OF_ENB2 + lsz(32, 16);
constexpr int OF_BBR2    = OF_BBR1 + lsz(95, 64);
constexpr int OF_BBB1    = OF_BBR2 + lsz(64, 32);
constexpr int OF_BBB2    = OF_BBB1 + lsz(95, 64);
constexpr int OF_LSTM_RUN = OF_BBB2 + lsz(64, 32);
constexpr int LSTM_SZ = 128 + 8 * 16 * 64 + 8 * 8 * 64;  // bias + ih + hh frags
constexpr int OF_LSTM_BLK = OF_LSTM_RUN + LSTM_SZ;
constexpr int WPK_TOTAL = OF_LSTM_BLK + LSTM_SZ;
constexpr int WPK_CAP = 65536;  // floats reserved at start of ws

// ---------------------------------------------------------------------------
// Weight packing. kind 0: bias (b [+ b2]); kind 1: fragment pack of w[O][I]:
//   dst[(nt*KT+kt)*64 + lane*2 + r] = W[nt*16 + (lane&15)][kt*4 + 2*(lane>>4)+r]
// ---------------------------------------------------------------------------
struct Job { const float* w; const float* b; const float* b2; int I, O, ofs, kind; };
struct JobTable { Job j[48]; int n; };

__global__ __launch_bounds__(256) void prep_kernel(JobTable T, float* __restrict__ dst) {
  Job jb = T.j[blockIdx.x];
  if (jb.kind == 0) {
    for (int c = threadIdx.x; c < jb.O; c += blockDim.x)
      dst[jb.ofs + c] = jb.b[c] + (jb.b2 ? jb.b2[c] : 0.f);
  } else {
    const int NT = jb.O >> 4, KT = (jb.I + 3) >> 2;
    const int tot = NT * KT * 64;
    for (int idx = threadIdx.x; idx < tot; idx += blockDim.x) {
      int r = idx & 1, lane = (idx >> 1) & 31, fk = idx >> 6;
      int nt = fk / KT, kt = fk - nt * KT;
      int col = nt * 16 + (lane & 15);
      int k = kt * 4 + 2 * (lane >> 4) + r;
      dst[jb.ofs + idx] = (k < jb.I) ? jb.w[col * jb.I + k] : 0.f;
    }
  }
}

// Y[16][O] = act(X[16][0..I) @ W^T + b). X row-major in LDS (pad cols zeroed).
// Lp = packed layer block: [bias O][frags].
template <int I, int O, bool RELU>
__device__ inline void lin16_lds(const float* __restrict__ X, int xp,
                                 const float* __restrict__ Lp,
                                 float* __restrict__ Y, int yp, int lane) {
  constexpr int NT = O / 16, KT = (I + 3) / 4;
  const int m = lane & 15, half = lane >> 4, n = lane & 15, kh = 2 * half;
  const float* Fp = Lp + NT * 16;
#pragma unroll
  for (int nt = 0; nt < NT; ++nt) {
    v8f acc;
    const float bv = Lp[nt * 16 + n];
#pragma unroll
    for (int r = 0; r < 8; ++r) acc[r] = bv;
#pragma unroll
    for (int kt = 0; kt < KT; ++kt) {
      v2f a = ld2(X + m * xp + kt * 4 + kh);
      v2f b = ld2(Fp + (nt * KT + kt) * 64 + lane * 2);
      acc = wmma4(a, b, acc);
    }
#pragma unroll
    for (int r = 0; r < 8; ++r) {
      const int row = r + 8 * half;
      float v = acc[r];
      if (RELU) v = fmaxf(v, 0.f);
      Y[row * yp + nt * 16 + n] = v;
    }
  }
}

// Same, D-fragments stored straight to global at G[row*gp + colofs + col].
template <int I, int O, bool RELU>
__device__ inline void lin16_glob(const float* __restrict__ X, int xp,
                                  const float* __restrict__ Lp,
                                  float* __restrict__ G, int gp, int colofs,
                                  int lane) {
  constexpr int NT = O / 16, KT = (I + 3) / 4;
  const int m = lane & 15, half = lane >> 4, n = lane & 15, kh = 2 * half;
  const float* Fp = Lp + NT * 16;
#pragma unroll
  for (int nt = 0; nt < NT; ++nt) {
    v8f acc;
    const float bv = Lp[nt * 16 + n];
#pragma unroll
    for (int r = 0; r < 8; ++r) acc[r] = bv;
#pragma unroll
    for (int kt = 0; kt < KT; ++kt) {
      v2f a = ld2(X + m * xp + kt * 4 + kh);
      v2f b = ld2(Fp + (nt * KT + kt) * 64 + lane * 2);
      acc = wmma4(a, b, acc);
    }
#pragma unroll
    for (int r = 0; r < 8; ++r) {
      const int row = r + 8 * half;
      float v = acc[r];
      if (RELU) v = fmaxf(v, 0.f);
      G[(size_t)row * gp + colofs + nt * 16 + n] = v;
    }
  }
}

// ---------------------------------------------------------------------------
// Kernel A: per-token encoder. 1 wave per WG, 16 tokens per wave.
// ---------------------------------------------------------------------------
__global__ __launch_bounds__(32) void encoder_kernel(
    const float* __restrict__ selfo, const float* __restrict__ tmo,
    const float* __restrict__ eno, const float* __restrict__ cpo,
    const float* __restrict__ mapo, const int* __restrict__ role,
    const float* __restrict__ mapin_w, const float* __restrict__ mapin_b,
    const float* __restrict__ role_emb, const float* __restrict__ pred_w,
    const float* __restrict__ pred_b, const float* __restrict__ ln1_g,
    const float* __restrict__ ln1_b, const float* __restrict__ ln2_g,
    const float* __restrict__ ln2_b, const float* __restrict__ wpk,
    float* __restrict__ lin_run, float* __restrict__ lin_blk,
    float* __restrict__ outp) {
  __shared__ float sX[16 * PX];
  __shared__ float sY[16 * PY];
  __shared__ float sC[16 * PC];
  __shared__ float sEB[16 * 18];
  const int lane = threadIdx.x;
  const int t0 = blockIdx.x * 16;

  // ---- pilot [self(15)|cp(10)] -> sX cols 0..24 (+zero pad 25..27); fs->cmd
  for (int i = lane; i < 16 * 15; i += 32) {
    int row = i / 15, c = i % 15;
    float v = selfo[(size_t)(t0 + row) * 15 + c];
    sX[row * PX + c] = v;
    sC[row * PC + c] = v;
  }
  for (int i = lane; i < 16 * 10; i += 32) {
    int row = i / 10, c = i % 10;
    sX[row * PX + 15 + c] = cpo[(size_t)(t0 + row) * 10 + c];
  }
  for (int i = lane; i < 16 * 3; i += 32) {  // pad to 28
    int row = i / 3, c = i % 3;
    sX[row * PX + 25 + c] = 0.f;
  }
  if (lane < 16) sC[lane * PC + 95] = 0.f;  // cmd pad col (95 -> 96)
  __syncthreads();
  lin16_lds<25, 64, true>(sX, PX, wpk + OF_PE_RUN1, sY, PY, lane);
  __syncthreads();
  lin16_glob<64, 32, true>(sY, PY, wpk + OF_PE_RUN2,
                           lin_run + (size_t)t0 * 64, 64, 0, lane);
  __syncthreads();
  lin16_lds<25, 64, true>(sX, PX, wpk + OF_PE_BLK1, sY, PY, lane);
  __syncthreads();
  lin16_glob<64, 32, true>(sY, PY, wpk + OF_PE_BLK2,
                           lin_blk + (size_t)t0 * 64, 64, 0, lane);
  __syncthreads();

  // ---- teammate latent -> cmd cols 15..30
  for (int i = lane; i < 16 * 14; i += 32) {
    int row = i / 14, c = i % 14;
    sX[row * PX + c] = tmo[(size_t)(t0 + row) * 14 + c];
  }
  {  // pad cols 14,15
    int row = lane >> 1, c = lane & 1;
    sX[row * PX + 14 + c] = 0.f;
  }
  __syncthreads();
  lin16_lds<14, 32, true>(sX, PX, wpk + OF_TM1, sY, PY, lane);
  __syncthreads();
  lin16_lds<32, 16, false>(sY, PY, wpk + OF_TM2, sC + 15, PC, lane);

  // ---- role embedding -> cmd cols 47..62
  for (int i = lane; i < 16 * 16; i += 32) {
    int row = i >> 4, c = i & 15;
    sC[row * PC + 47 + c] = role_emb[(size_t)role[t0 + row] * 16 + c];
  }
  __syncthreads();

  // ---- enemies: 32 rows (16 tok x 2) as two 16-row halves
  for (int hlf = 0; hlf < 2; ++hlf) {
    for (int i = lane; i < 16 * 14; i += 32) {
      int row = i / 14, c = i % 14;
      int tok = hlf * 8 + (row >> 1), e = row & 1;
      sX[row * PX + c] = eno[((size_t)(t0 + tok) * 2 + e) * 14 + c];
    }
    {
      int row = lane >> 1, c = lane & 1;
      sX[row * PX + 14 + c] = 0.f;
    }
    __syncthreads();
    lin16_lds<14, 32, true>(sX, PX, wpk + OF_ENR1, sY, PY, lane);
    __syncthreads();
    lin16_lds<32, 16, false>(sY, PY, wpk + OF_ENR2, sX + 32, PX, lane);
    __syncthreads();
    lin16_lds<14, 32, true>(sX, PX, wpk + OF_ENB1, sY, PY, lane);
    __syncthreads();
    lin16_lds<32, 16, false>(sY, PY, wpk + OF_ENB2, sX + 48, PX, lane);
    __syncthreads();
    {  // pred = (en_run+en_blk)/2 @ predW^T + pred_b
      int row = lane >> 1, o = lane & 1;
      float s = pred_b[o];
      for (int j = 0; j < 16; ++j)
        s += pred_w[o * 16 + j] * 0.5f *
             (sX[row * PX + 32 + j] + sX[row * PX + 48 + j]);
      int tok = hlf * 8 + (row >> 1), e = row & 1;
      outp[(size_t)8 * NBS + (size_t)(t0 + tok) * 4 + e * 2 + o] = s;
    }
    for (int i = lane; i < 8 * 16; i += 32) {  // ctx max over enemies
      int tk = i >> 4, c = i & 15;
      int r0 = tk * 2, r1 = tk * 2 + 1;
      sC[(hlf * 8 + tk) * PC + 31 + c] =
          fmaxf(sX[r0 * PX + 32 + c], sX[r1 * PX + 32 + c]);
      sEB[(hlf * 8 + tk) * 18 + c] =
          fmaxf(sX[r0 * PX + 48 + c], sX[r1 * PX + 48 + c]);
    }
    __syncthreads();
  }

  // ---- map encoder: 2 tokens (16 rows = 2x8 cp) per chunk -> cmd 63..94
  for (int ch = 0; ch < 8; ++ch) {
    for (int i = lane; i < 16 * 32; i += 32) {  // x0 = relu(map_in)
      int row = i >> 5, c = i & 31;
      int tok = ch * 2 + (row >> 3), cp = row & 7;
      const float* mo = &mapo[((size_t)(t0 + tok) * 8 + cp) * 2];
      sX[row * PX + c] =
          fmaxf(mapin_w[c * 2 + 0] * mo[0] + mapin_w[c * 2 + 1] * mo[1] +
                    mapin_b[c],
                0.f);
    }
    __syncthreads();
    lin16_lds<32, 96, false>(sX, PX, wpk + OF_QKV, sY, PY, lane);
    __syncthreads();
    {  // attention: lane = q(0..7) + head*8 + tok*16 ; o -> sX cols 32..63
      int q = lane & 7, hd = (lane >> 3) & 1, tk = lane >> 4;
      int qrow = tk * 8 + q;
      float sc[8], mx = -1e30f;
      for (int kk = 0; kk < 8; ++kk) {
        float s = 0.f;
        int krow = tk * 8 + kk;
        for (int d = 0; d < 16; ++d)
          s += sY[qrow * PY + hd * 16 + d] * sY[krow * PY + 32 + hd * 16 + d];
        s *= 0.25f;
        sc[kk] = s;
        mx = fmaxf(mx, s);
      }
      float den = 0.f;
      for (int kk = 0; kk < 8; ++kk) {
        sc[kk] = expf(sc[kk] - mx);
        den += sc[kk];
      }
      float inv = 1.f / den;
      for (int d = 0; d < 16; ++d) {
        float o = 0.f;
        for (int kk = 0; kk < 8; ++kk)
          o += sc[kk] * sY[(tk * 8 + kk) * PY + 64 + hd * 16 + d];
        sX[qrow * PX + 32 + hd * 16 + d] = o * inv;
      }
    }
    __syncthreads();
    lin16_lds<32, 32, false>(sX + 32, PX, wpk + OF_AOUT, sX + 64, PX, lane);
    __syncthreads();
    for (int i = lane; i < 16 * 32; i += 32) {  // residual 1
      int row = i >> 5, c = i & 31;
      sX[row * PX + 64 + c] += sX[row * PX + c];
    }
    __syncthreads();
    if (lane < 16) {  // LN1 in place (cols 64..95)
      float m = 0.f;
      for (int c = 0; c < 32; ++c) m += sX[lane * PX + 64 + c];
      m *= (1.f / 32.f);
      float v = 0.f;
      for (int c = 0; c < 32; ++c) {
        float d = sX[lane * PX + 64 + c] - m;
        v += d * d;
      }
      v *= (1.f / 32.f);
      float is = rsqrtf(v + 1e-5f);
      for (int c = 0; c < 32; ++c) {
        float d = (sX[lane * PX + 64 + c] - m) * is;
        sX[lane * PX + 64 + c] = d * ln1_g[c] + ln1_b[c];
      }
    }
    __syncthreads();
    lin16_lds<32, 64, true>(sX + 64, PX, wpk + OF_FF1, sY, PY, lane);
    __syncthreads();
    lin16_lds<64, 32, false>(sY, PY, wpk + OF_FF2, sX, PX, lane);
    __syncthreads();
    if (lane < 16) {  // residual 2 + LN2 -> cols 0..31
      float tv[32];
      float m = 0.f;
      for (int c = 0; c < 32; ++c) {
        tv[c] = sX[lane * PX + c] + sX[lane * PX + 64 + c];
        m += tv[c];
      }
      m *= (1.f / 32.f);
      float v = 0.f;
      for (int c = 0; c < 32; ++c) {
        float d = tv[c] - m;
        v += d * d;
      }
      v *= (1.f / 32.f);
      float is = rsqrtf(v + 1e-5f);
      for (int c = 0; c < 32; ++c)
        sX[lane * PX + c] = (tv[c] - m) * is * ln2_g[c] + ln2_b[c];
    }
    __syncthreads();
    for (int i = lane; i < 2 * 32; i += 32) {  // max over 8 cp -> cmd
      int tk = i >> 5, c = i & 31;
      float mv = -1e30f;
      for (int cp = 0; cp < 8; ++cp) mv = fmaxf(mv, sX[(tk * 8 + cp) * PX + c]);
      sC[(ch * 2 + tk) * PC + 63 + c] = mv;
    }
    __syncthreads();
  }

  // ---- backbones on cmd [16][95]
  lin16_lds<95, 64, true>(sC, PC, wpk + OF_BBR1, sY, PY, lane);
  __syncthreads();
  lin16_glob<64, 32, true>(sY, PY, wpk + OF_BBR2,
                           lin_run + (size_t)t0 * 64, 64, 32, lane);
  __syncthreads();
  for (int i = lane; i < 16 * 16; i += 32) {  // swap ctx -> blk
    int row = i >> 4, c = i & 15;
    sC[row * PC + 31 + c] = sEB[row * 18 + c];
  }
  __syncthreads();
  lin16_lds<95, 64, true>(sC, PC, wpk + OF_BBB1, sY, PY, lane);
  __syncthreads();
  lin16_glob<64, 32, true>(sY, PY, wpk + OF_BBB2,
                           lin_blk + (size_t)t0 * 64, 64, 32, lane);
}

// ---------------------------------------------------------------------------
// Kernel B: LSTM. 1 wave per (16-batch tile, run/blk); packed frags in LDS.
// ---------------------------------------------------------------------------
__global__ __launch_bounds__(32) void lstm_kernel(
    const float* __restrict__ lin_base, const float* __restrict__ wpk,
    float* __restrict__ hout_base, float* __restrict__ outp) {
  __shared__ float sF[LSTM_SZ - 128];  // ih frags [0,8192) | hh frags [8192,12288)
  __shared__ float sH[16 * 34];
  const int typ = blockIdx.x & 1;
  const int b0 = (blockIdx.x >> 1) * 16;
  const float* lin = lin_base + (size_t)typ * NBS * 64;
  const float* Lp = wpk + OF_LSTM_RUN + (size_t)typ * LSTM_SZ;
  float* hout = hout_base + (size_t)typ * NBS * 32;
  const int lane = threadIdx.x;
  const int m = lane & 15, half = lane >> 4, n = lane & 15, kh = 2 * half;

  for (int i = lane; i < LSTM_SZ - 128; i += 32) sF[i] = Lp[128 + i];
  for (int i = lane; i < 16 * 32; i += 32) {
    int r = i >> 5, c = i & 31;
    sH[r * 34 + c] = 0.f;
  }
  float bias[8];
#pragma unroll
  for (int nt = 0; nt < 8; ++nt) bias[nt] = Lp[nt * 16 + n];
  __syncthreads();

  v8f c0 = {}, c1 = {}, hl0 = {}, hl1 = {};
#pragma unroll 1
  for (int t = 0; t < NS; ++t) {
    v8f acc[8];
#pragma unroll
    for (int nt = 0; nt < 8; ++nt)
#pragma unroll
      for (int r = 0; r < 8; ++r) acc[nt][r] = bias[nt];

    const size_t lbase = ((size_t)(b0 + m) * NS + t) * 64;
#pragma unroll
    for (int kt = 0; kt < 16; ++kt) {  // input projection (A from global lin)
      v2f a = ld2(lin + lbase + kt * 4 + kh);
#pragma unroll
      for (int nt = 0; nt < 8; ++nt) {
        v2f b = ld2(sF + (nt * 16 + kt) * 64 + lane * 2);
        acc[nt] = wmma4(a, b, acc[nt]);
      }
    }
#pragma unroll
    for (int kt = 0; kt < 8; ++kt) {  // recurrent projection (A from LDS h)
      v2f a = ld2(sH + m * 34 + kt * 4 + kh);
#pragma unroll
      for (int nt = 0; nt < 8; ++nt) {
        v2f b = ld2(sF + 8192 + (nt * 8 + kt) * 64 + lane * 2);
        acc[nt] = wmma4(a, b, acc[nt]);
      }
    }
    __syncthreads();
#pragma unroll
    for (int ct = 0; ct < 2; ++ct) {  // gates i,f,g,o = ntiles {ct,ct+2,ct+4,ct+6}
      v8f& cc = ct ? c1 : c0;
      v8f& hl = ct ? hl1 : hl0;
#pragma unroll
      for (int r = 0; r < 8; ++r) {
        float ig = sigm(acc[ct][r]);
        float fg = sigm(acc[ct + 2][r]);
        float gg = tanhf(acc[ct + 4][r]);
        float og = sigm(acc[ct + 6][r]);
        float cn = fg * cc[r] + ig * gg;
        float hv = og * tanhf(cn);
        cc[r] = cn;
        hl[r] = hv;
        const int row = r + 8 * half;
        sH[row * 34 + ct * 16 + n] = hv;
        hout[((size_t)(b0 + row) * NS + t) * 32 + ct * 16 + n] = hv;
      }
    }
    __syncthreads();
  }
  const size_t hn_off = (size_t)12 * NBS + (typ ? (size_t)64 * NB : 0);
  const size_t cn_off = hn_off + (size_t)32 * NB;
#pragma unroll
  for (int ct = 0; ct < 2; ++ct) {
    const v8f& cc = ct ? c1 : c0;
    const v8f& hl = ct ? hl1 : hl0;
#pragma unroll
    for (int r = 0; r < 8; ++r) {
      const int row = r + 8 * half;
      outp[hn_off + (size_t)(b0 + row) * 32 + ct * 16 + n] = hl[r];
      outp[cn_off + (size_t)(b0 + row) * 32 + ct * 16 + n] = cc[r];
    }
  }
}

// ---------------------------------------------------------------------------
// Kernel C: role-mask fuse + output heads (one lane per token).
// ---------------------------------------------------------------------------
__global__ __launch_bounds__(256) void heads_kernel(
    const float* __restrict__ hr, const float* __restrict__ hb,
    const int* __restrict__ role, const float* __restrict__ wt,
    const float* __restrict__ bt, const float* __restrict__ wa,
    const float* __restrict__ ba, const float* __restrict__ wsh,
    const float* __restrict__ bsh, const float* __restrict__ wbo,
    const float* __restrict__ bbo, const float* __restrict__ logstd,
    float* __restrict__ outp) {
  const int tok = blockIdx.x * blockDim.x + threadIdx.x;
  if (tok >= NBS) return;
  const float mask = (float)role[tok];
  float h[32];
  const float* a = hr + (size_t)tok * 32;
  const float* b = hb + (size_t)tok * 32;
  for (int c = 0; c < 32; ++c) h[c] = a[c] * (1.f - mask) + b[c] * mask;
  float th = bt[0], an = ba[0];
  float s0 = bsh[0], s1 = bsh[1], o0 = bbo[0], o1 = bbo[1];
  for (int c = 0; c < 32; ++c) {
    th += wt[c] * h[c];
    an += wa[c] * h[c];
    s0 += wsh[c] * h[c];
    s1 += wsh[32 + c] * h[c];
    o0 += wbo[c] * h[c];
    o1 += wbo[32 + c] * h[c];
  }
  outp[(size_t)tok * 2 + 0] = sigm(th);
  outp[(size_t)tok * 2 + 1] = tanhf(an);
  outp[(size_t)2 * NBS + tok * 2 + 0] = fmaxf(expf(logstd[0]), 0.05f);
  outp[(size_t)2 * NBS + tok * 2 + 1] = fmaxf(expf(logstd[1]), 0.05f);
  outp[(size_t)4 * NBS + tok * 2 + 0] = s0;
  outp[(size_t)4 * NBS + tok * 2 + 1] = s1;
  outp[(size_t)6 * NBS + tok * 2 + 0] = o0;
  outp[(size_t)6 * NBS + tok * 2 + 1] = o1;
}

// ---------------------------------------------------------------------------
extern "C" void kernel_launch(void* const* d_in, const int* in_sizes, int n_in,
                              void* d_out, int out_size, void* d_ws,
                              size_t ws_size, hipStream_t stream) {
  const float** q = (const float**)d_in;
  const float* selfo = q[0];
  const float* tmo = q[1];
  const float* eno = q[2];
  const float* cpo = q[3];
  const float* mapo = q[4];
  const int* role = (const int*)d_in[5];

  // Param leaves in dict-insertion order, {w, b} per Linear.
  int i = 6;
  const float *pe_run1_w = q[i++], *pe_run1_b = q[i++];
  const float *pe_run2_w = q[i++], *pe_run2_b = q[i++];
  const float *pe_blk1_w = q[i++], *pe_blk1_b = q[i++];
  const float *pe_blk2_w = q[i++], *pe_blk2_b = q[i++];
  const float *map_in_w = q[i++], *map_in_b = q[i++];
  const float *attn_qkv_w = q[i++], *attn_qkv_b = q[i++];
  const float *attn_out_w = q[i++], *attn_out_b = q[i++];
  const float *ln1_g = q[i++], *ln1_b = q[i++];
  const float *ff1_w = q[i++], *ff1_b = q[i++];
  const float *ff2_w = q[i++], *ff2_b = q[i++];
  const float *ln2_g = q[i++], *ln2_b = q[i++];
  const float *role_emb = q[i++];
  const float *tm1_w = q[i++], *tm1_b = q[i++];
  const float *tm2_w = q[i++], *tm2_b = q[i++];
  const float *en_run1_w = q[i++], *en_run1_b = q[i++];
  const float *en_run2_w = q[i++], *en_run2_b = q[i++];
  const float *en_blk1_w = q[i++], *en_blk1_b = q[i++];
  const float *en_blk2_w = q[i++], *en_blk2_b = q[i++];
  const float *pred_w = q[i++], *pred_b = q[i++];
  const float *bb_run1_w = q[i++], *bb_run1_b = q[i++];
  const float *bb_run2_w = q[i++], *bb_run2_b = q[i++];
  const float *bb_blk1_w = q[i++], *bb_blk1_b = q[i++];
  const float *bb_blk2_w = q[i++], *bb_blk2_b = q[i++];
  const float *lstm_run_ih_w = q[i++], *lstm_run_ih_b = q[i++];
  const float *lstm_run_hh_w = q[i++], *lstm_run_hh_b = q[i++];
  const float *lstm_blk_ih_w = q[i++], *lstm_blk_ih_b = q[i++];
  const float *lstm_blk_hh_w = q[i++], *lstm_blk_hh_b = q[i++];
  const float *h_thrust_w = q[i++], *h_thrust_b = q[i++];
  const float *h_angle_w = q[i++], *h_angle_b = q[i++];
  const float *h_shield_w = q[i++], *h_shield_b = q[i++];
  const float *h_boost_w = q[i++], *h_boost_b = q[i++];
  const float *logstd = q[i++];

  // ws: [wpk WPK_CAP][lin_run NBS*64][lin_blk NBS*64][hout_run NBS*32][hout_blk NBS*32]
  float* wsf = (float*)d_ws;
  float* wpk = wsf;
  float* lin_run = wsf + WPK_CAP;
  float* lin_blk = lin_run + (size_t)NBS * 64;
  float* hout_run = lin_blk + (size_t)NBS * 64;
  float* hout_blk = hout_run + (size_t)NBS * 32;
  float* outp = (float*)d_out;

  // ---- build pack jobs
  JobTable T;
  T.n = 0;
  auto add = [&](const float* w, const float* b, int I, int O, int ofs) {
    T.j[T.n++] = {nullptr, b, nullptr, I, O, ofs, 0};
    T.j[T.n++] = {w, nullptr, nullptr, I, O, ofs + O, 1};
  };
  add(pe_run1_w, pe_run1_b, 25, 64, OF_PE_RUN1);
  add(pe_run2_w, pe_run2_b, 64, 32, OF_PE_RUN2);
  add(pe_blk1_w, pe_blk1_b, 25, 64, OF_PE_BLK1);
  add(pe_blk2_w, pe_blk2_b, 64, 32, OF_PE_BLK2);
  add(attn_qkv_w, attn_qkv_b, 32, 96, OF_QKV);
  add(attn_out_w, attn_out_b, 32, 32, OF_AOUT);
  add(ff1_w, ff1_b, 32, 64, OF_FF1);
  add(ff2_w, ff2_b, 64, 32, OF_FF2);
  add(tm1_w, tm1_b, 14, 32, OF_TM1);
  add(tm2_w, tm2_b, 32, 16, OF_TM2);
  add(en_run1_w, en_run1_b, 14, 32, OF_ENR1);
  add(en_run2_w, en_run2_b, 32, 16, OF_ENR2);
  add(en_blk1_w, en_blk1_b, 14, 32, OF_ENB1);
  add(en_blk2_w, en_blk2_b, 32, 16, OF_ENB2);
  add(bb_run1_w, bb_run1_b, 95, 64, OF_BBR1);
  add(bb_run2_w, bb_run2_b, 64, 32, OF_BBR2);
  add(bb_blk1_w, bb_blk1_b, 95, 64, OF_BBB1);
  add(bb_blk2_w, bb_blk2_b, 64, 32, OF_BBB2);
  auto addlstm = [&](const float* wih, const float* bih, const float* whh,
                     const float* bhh, int ofs) {
    T.j[T.n++] = {nullptr, bih, bhh, 0, 128, ofs, 0};
    T.j[T.n++] = {wih, nullptr, nullptr, 64, 128, ofs + 128, 1};
    T.j[T.n++] = {whh, nullptr, nullptr, 32, 128, ofs + 128 + 8192, 1};
  };
  addlstm(lstm_run_ih_w, lstm_run_ih_b, lstm_run_hh_w, lstm_run_hh_b, OF_LSTM_RUN);
  addlstm(lstm_blk_ih_w, lstm_blk_ih_b, lstm_blk_hh_w, lstm_blk_hh_b, OF_LSTM_BLK);

  prep_kernel<<<T.n, 256, 0, stream>>>(T, wpk);
  encoder_kernel<<<NBS / 16, 32, 0, stream>>>(
      selfo, tmo, eno, cpo, mapo, role, map_in_w, map_in_b, role_emb, pred_w,
      pred_b, ln1_g, ln1_b, ln2_g, ln2_b, wpk, lin_run, lin_blk, outp);
  lstm_kernel<<<(NB / 16) * 2, 32, 0, stream>>>(lin_run, wpk, hout_run, outp);
  heads_kernel<<<NBS / 256, 256, 0, stream>>>(
      hout_run, hout_blk, role, h_thrust_w, h_thrust_b, h_angle_w, h_angle_b,
      h_shield_w, h_shield_b, h_boost_w, h_boost_b, logstd, outp);
}